// GNNClassifier_88648124990131
// MI455X (gfx1250) — compile-verified
//
#include <hip/hip_runtime.h>
#include <hip/hip_bf16.h>

typedef float v2f __attribute__((ext_vector_type(2)));
typedef float v8f __attribute__((ext_vector_type(8)));

#define SCAN_CHUNK 256

// ---------------------------------------------------------------------------
// Utility kernels
// ---------------------------------------------------------------------------
__global__ void fill_zero_f32_kernel(float* __restrict__ p, size_t count) {
    size_t i = (size_t)blockIdx.x * blockDim.x + threadIdx.x;
    size_t stride = (size_t)gridDim.x * blockDim.x;
    for (; i < count; i += stride) p[i] = 0.0f;
}

__global__ void fill_zero_i32_kernel(int* __restrict__ p, size_t count) {
    size_t i = (size_t)blockIdx.x * blockDim.x + threadIdx.x;
    size_t stride = (size_t)gridDim.x * blockDim.x;
    for (; i < count; i += stride) p[i] = 0;
}

// deg[dst] += 1 for every edge (int counts)
__global__ void edge_deg_kernel(const int* __restrict__ ei, int E,
                                int* __restrict__ deg) {
    int i = blockIdx.x * blockDim.x + threadIdx.x;
    if (i < E) atomicAdd(&deg[ei[E + i]], 1);      // row 1 of edge_index = dst
}

// gcnt[batch[i]] += 1 for every node
__global__ void node_cnt_kernel(const int* __restrict__ batch, int N,
                                float* __restrict__ gcnt) {
    int i = blockIdx.x * blockDim.x + threadIdx.x;
    if (i < N) atomicAdd(&gcnt[batch[i]], 1.0f);
}

// ---------------------------------------------------------------------------
// 3-kernel exclusive prefix sum over deg -> rowptr (CSR by dst).
// Supports N <= SCAN_CHUNK * 1024 = 262144.
// ---------------------------------------------------------------------------
__global__ void scan_block_kernel(const int* __restrict__ deg, int N,
                                  int* __restrict__ incl, int* __restrict__ bsum) {
    __shared__ int sh[SCAN_CHUNK];
    const int t = threadIdx.x;
    const int i = blockIdx.x * SCAN_CHUNK + t;
    sh[t] = (i < N) ? deg[i] : 0;
    __syncthreads();
    for (int ofs = 1; ofs < SCAN_CHUNK; ofs <<= 1) {
        int add = (t >= ofs) ? sh[t - ofs] : 0;
        __syncthreads();
        sh[t] += add;
        __syncthreads();
    }
    if (i < N) incl[i] = sh[t];
    if (t == SCAN_CHUNK - 1) bsum[blockIdx.x] = sh[t];
}

// single block, 1024 threads: bsum <- exclusive scan of bsum[0..nb)
__global__ void scan_sums_kernel(int* __restrict__ bsum, int nb) {
    __shared__ int sh[1024];
    const int t = threadIdx.x;
    sh[t] = (t < nb) ? bsum[t] : 0;
    __syncthreads();
    for (int ofs = 1; ofs < 1024; ofs <<= 1) {
        int add = (t >= ofs) ? sh[t - ofs] : 0;
        __syncthreads();
        sh[t] += add;
        __syncthreads();
    }
    if (t < nb) bsum[t] = (t == 0) ? 0 : sh[t - 1];
}

// rowptr[i+1] = incl[i] + bsum_excl[i / SCAN_CHUNK]; rowptr[0] = 0
__global__ void finalize_rowptr_kernel(const int* __restrict__ incl,
                                       const int* __restrict__ bsum_excl,
                                       int N, int* __restrict__ rowptr) {
    int i = blockIdx.x * blockDim.x + threadIdx.x;
    if (i < N) rowptr[i + 1] = incl[i] + bsum_excl[i / SCAN_CHUNK];
    if (i == 0) rowptr[0] = 0;
}

__global__ void copy_i32_kernel(const int* __restrict__ src,
                                int* __restrict__ dst, int n) {
    int i = blockIdx.x * blockDim.x + threadIdx.x;
    if (i < n) dst[i] = src[i];
}

// col[cursor[dst]++] = src  (counting-sort fill; order within a row arbitrary)
__global__ void csr_fill_kernel(const int* __restrict__ ei, int E,
                                int* __restrict__ cursor, int* __restrict__ col) {
    int e = blockIdx.x * blockDim.x + threadIdx.x;
    if (e < E) {
        int t = ei[E + e];
        int pos = atomicAdd(&cursor[t], 1);
        col[pos] = ei[e];
    }
}

// h0[n, :] = emb[x[n], :]   (float4 granularity)
__global__ void embed_gather_kernel(const int* __restrict__ x,
                                    const float* __restrict__ emb,
                                    float* __restrict__ h0,
                                    size_t total_chunks, int cshift) {
    size_t i = (size_t)blockIdx.x * blockDim.x + threadIdx.x;
    size_t stride = (size_t)gridDim.x * blockDim.x;
    const float4* esrc = (const float4*)emb;
    float4* dst = (float4*)h0;
    const int cmask = (1 << cshift) - 1;
    for (; i < total_chunks; i += stride) {
        size_t node = i >> cshift;
        int c = (int)(i & cmask);
        dst[i] = esrc[((size_t)x[node] << cshift) + c];
    }
}

// agg[i, d] = mean over incoming edges of h[src, d]  (gather form, no atomics)
__global__ void gather_mean_kernel(const int* __restrict__ rowptr,
                                   const int* __restrict__ col,
                                   const float* __restrict__ h, int D,
                                   float* __restrict__ agg) {
    const int i = blockIdx.x;        // dst node
    const int d = threadIdx.x;       // feature
    const int beg = rowptr[i];
    const int end = rowptr[i + 1];
    float s = 0.0f;
    for (int e = beg; e < end; ++e) {
        if (e + 1 < end)             // prefetch next neighbor row (global_prefetch_b8)
            __builtin_prefetch(&h[(size_t)col[e + 1] * D + d], 0, 0);
        s += h[(size_t)col[e] * D + d];
    }
    const float cnt = (float)(end - beg);
    agg[(size_t)i * D + d] = s / fmaxf(cnt, 1.0f);
}

// ---------------------------------------------------------------------------
// SAGE layer GEMM:  out = relu(A1 @ Wl + A2 @ Wr + bias)
//   A1, A2: [N_rows, K] (ldA1/ldA2), Wl, Wr: [K, HIDp] row-major
//   FUSE_POOL=false: store h1[row, n]
//   FUSE_POOL=true : atomicAdd into gacc[batch[row], n]  (mean-pool numerator)
// One wave32 computes one 16x16 tile via V_WMMA_F32_16X16X4_F32.
// A fragment (16x4 f32): lane 0-15 = M rows, v[0]=K{0|2}, v[1]=K{1|3}
// B fragment (4x16 f32): lane 0-15 = N cols, mirrored K packing.
// ---------------------------------------------------------------------------
template <bool FUSE_POOL>
__global__ void sage_gemm_kernel(const float* __restrict__ A1, int ldA1,
                                 const float* __restrict__ A2, int ldA2,
                                 const float* __restrict__ Wl,
                                 const float* __restrict__ Wr,
                                 const float* __restrict__ bias,
                                 int K, int N_rows, int HIDp,
                                 float* __restrict__ out,
                                 const int* __restrict__ batch) {
    const int lane  = threadIdx.x & 31;
    const int wave  = threadIdx.x >> 5;
    const int mtile = blockIdx.x;
    const int ntile = blockIdx.y * (blockDim.x >> 5) + wave;
    const int n     = ntile * 16 + (lane & 15);

    const int mr = mtile * 16 + (lane & 15);
    const int mc = mr < N_rows ? mr : N_rows - 1;   // clamp (stores predicated)
    const int koff = (lane >> 4) << 1;              // 0 for lanes 0-15, 2 for 16-31

    const float* a1row = A1 + (size_t)mc * ldA1 + koff;
    const float* a2row = A2 + (size_t)mc * ldA2 + koff;

    v8f acc = {};
#pragma unroll 4
    for (int kk = 0; kk < K; kk += 4) {
        v2f a = *(const v2f*)(a1row + kk);
        v2f b;
        b.x = Wl[(size_t)(kk + koff) * HIDp + n];
        b.y = Wl[(size_t)(kk + koff + 1) * HIDp + n];
        acc = __builtin_amdgcn_wmma_f32_16x16x4_f32(false, a, false, b,
                                                    (short)0, acc, false, false);
    }
#pragma unroll 4
    for (int kk = 0; kk < K; kk += 4) {
        v2f a = *(const v2f*)(a2row + kk);
        v2f b;
        b.x = Wr[(size_t)(kk + koff) * HIDp + n];
        b.y = Wr[(size_t)(kk + koff + 1) * HIDp + n];
        acc = __builtin_amdgcn_wmma_f32_16x16x4_f32(false, a, false, b,
                                                    (short)0, acc, false, false);
    }

    // Epilogue: bias + relu + store. C/D layout: VGPR r -> M = r (+8 hi lanes).
    const float bn = bias[n];
    const int mbase = mtile * 16 + ((lane >> 4) << 3);
#pragma unroll
    for (int r = 0; r < 8; ++r) {
        int row = mbase + r;
        if (row < N_rows) {
            float v = acc[r] + bn;
            v = v > 0.0f ? v : 0.0f;
            if (FUSE_POOL) {
                atomicAdd(&out[(size_t)batch[row] * HIDp + n], v);
            } else {
                out[(size_t)row * HIDp + n] = v;
            }
        }
    }
}

// ---------------------------------------------------------------------------
// Classifier:  out[g, c] = (gacc[g, :] / max(gcnt[g],1)) @ Wo + bo
// NCLS (=10) padded to one 16-wide WMMA tile; pad columns masked to zero.
// ---------------------------------------------------------------------------
__global__ void classifier_gemm_kernel(const float* __restrict__ gacc,
                                       const float* __restrict__ gcnt,
                                       const float* __restrict__ Wo,
                                       const float* __restrict__ bo,
                                       int G, int HIDp, int NC,
                                       float* __restrict__ out) {
    const int lane  = threadIdx.x & 31;
    const int wave  = threadIdx.x >> 5;
    const int mtile = blockIdx.x * (blockDim.x >> 5) + wave;
    const int n     = lane & 15;

    const int mr = mtile * 16 + (lane & 15);
    const int mc = mr < G ? mr : G - 1;
    const float inv = 1.0f / fmaxf(gcnt[mc], 1.0f);
    const int koff = (lane >> 4) << 1;

    const int   ncol  = n < NC ? n : NC - 1;   // clamp to stay in bounds
    const float nmask = n < NC ? 1.0f : 0.0f;  // zero out padded columns

    const float* arow = gacc + (size_t)mc * HIDp + koff;

    v8f acc = {};
#pragma unroll 4
    for (int kk = 0; kk < HIDp; kk += 4) {
        v2f a;
        a.x = arow[kk] * inv;
        a.y = arow[kk + 1] * inv;
        v2f b;
        b.x = Wo[(size_t)(kk + koff) * NC + ncol] * nmask;
        b.y = Wo[(size_t)(kk + koff + 1) * NC + ncol] * nmask;
        acc = __builtin_amdgcn_wmma_f32_16x16x4_f32(false, a, false, b,
                                                    (short)0, acc, false, false);
    }

    if (n < NC) {
        const float bn = bo[n];
        const int mbase = mtile * 16 + ((lane >> 4) << 3);
#pragma unroll
        for (int r = 0; r < 8; ++r) {
            int row = mbase + r;
            if (row < G) out[(size_t)row * NC + n] = acc[r] + bn;
        }
    }
}

// ---------------------------------------------------------------------------
// Host launcher
// ---------------------------------------------------------------------------
extern "C" void kernel_launch(void* const* d_in, const int* in_sizes, int n_in,
                              void* d_out, int out_size, void* d_ws, size_t ws_size,
                              hipStream_t stream) {
    (void)n_in; (void)ws_size;
    const int*   x     = (const int*)d_in[0];
    const int*   ei    = (const int*)d_in[1];
    const int*   batch = (const int*)d_in[2];
    const float* emb   = (const float*)d_in[3];
    const float* W1l   = (const float*)d_in[4];
    const float* W1r   = (const float*)d_in[5];
    const float* b1    = (const float*)d_in[6];
    const float* W2l   = (const float*)d_in[7];
    const float* W2r   = (const float*)d_in[8];
    const float* b2    = (const float*)d_in[9];
    const float* Wo    = (const float*)d_in[10];
    const float* bo    = (const float*)d_in[11];
    float* out = (float*)d_out;

    const int N    = in_sizes[0];            // 50000
    const int E    = in_sizes[1] / 2;        // 800000
    const int HID  = in_sizes[6];            // 256 (b1)
    const int EMBd = in_sizes[4] / HID;      // 128 (W1l is EMB x HID)
    const int NC   = in_sizes[11];           // 10  (bo)
    const int G    = out_size / NC;          // 256

    const int nscan = (N + SCAN_CHUNK - 1) / SCAN_CHUNK;   // <= 1024

    // Workspace layout: int region first, then float region (16B aligned).
    int* wsi = (int*)d_ws;
    size_t ioff = 0;
    int* deg    = wsi + ioff; ioff += (size_t)N;
    int* incl   = wsi + ioff; ioff += (size_t)N;       // block-inclusive scan
    int* bsum   = wsi + ioff; ioff += 1024;            // block sums
    int* rowptr = wsi + ioff; ioff += (size_t)N + 1;
    int* cursor = wsi + ioff; ioff += (size_t)N;
    int* col    = wsi + ioff; ioff += (size_t)E;
    ioff = (ioff + 3) & ~(size_t)3;                    // 16B align float region

    float* wsf = (float*)(wsi + ioff);
    size_t foff = 0;
    float* gcnt = wsf + foff; foff += (size_t)G;
    float* gacc = wsf + foff; foff += (size_t)G * HID;
    foff = (foff + 3) & ~(size_t)3;
    float* h0   = wsf + foff; foff += (size_t)N * EMBd;
    float* agg  = wsf + foff; foff += (size_t)N * HID;
    float* h1   = wsf + foff; foff += (size_t)N * HID;

    const int waves_per_block = 8;           // 256 threads

    // 1) zero small accumulators (deg; gcnt+gacc contiguous)
    fill_zero_i32_kernel<<<256, 256, 0, stream>>>(deg, (size_t)N);
    fill_zero_f32_kernel<<<256, 256, 0, stream>>>(gcnt, (size_t)G + (size_t)G * HID);

    // 2) degree / graph counts
    edge_deg_kernel<<<(E + 255) / 256, 256, 0, stream>>>(ei, E, deg);
    node_cnt_kernel<<<(N + 255) / 256, 256, 0, stream>>>(batch, N, gcnt);

    // 3) CSR-by-dst build: scan -> rowptr -> cursor -> bucket fill
    scan_block_kernel<<<nscan, SCAN_CHUNK, 0, stream>>>(deg, N, incl, bsum);
    scan_sums_kernel<<<1, 1024, 0, stream>>>(bsum, nscan);
    finalize_rowptr_kernel<<<(N + 255) / 256, 256, 0, stream>>>(incl, bsum, N, rowptr);
    copy_i32_kernel<<<(N + 255) / 256, 256, 0, stream>>>(rowptr, cursor, N);
    csr_fill_kernel<<<(E + 255) / 256, 256, 0, stream>>>(ei, E, cursor, col);

    // 4) embedding gather
    {
        int chunks = EMBd >> 2;              // float4 chunks per row
        int cshift = __builtin_ctz(chunks);
        embed_gather_kernel<<<2048, 256, 0, stream>>>(
            x, emb, h0, (size_t)N * chunks, cshift);
    }

    // 5) layer-1 neighbor mean (gather; fused /max(deg,1), no atomics, no zero-fill)
    gather_mean_kernel<<<N, EMBd, 0, stream>>>(rowptr, col, h0, EMBd, agg);

    // 6) layer-1 WMMA GEMM -> h1
    {
        dim3 grid((N + 15) / 16, (HID / 16) / waves_per_block);
        sage_gemm_kernel<false><<<grid, 32 * waves_per_block, 0, stream>>>(
            agg, EMBd, h0, EMBd, W1l, W1r, b1, EMBd, N, HID, h1, nullptr);
    }

    // 7) layer-2 neighbor mean
    gather_mean_kernel<<<N, HID, 0, stream>>>(rowptr, col, h1, HID, agg);

    // 8) layer-2 WMMA GEMM, fused relu + mean-pool accumulation into gacc
    {
        dim3 grid((N + 15) / 16, (HID / 16) / waves_per_block);
        sage_gemm_kernel<true><<<grid, 32 * waves_per_block, 0, stream>>>(
            agg, HID, h1, HID, W2l, W2r, b2, HID, N, HID, gacc, batch);
    }

    // 9) classifier WMMA GEMM (divides by graph node counts internally)
    {
        int mtiles = (G + 15) / 16;
        classifier_gemm_kernel<<<(mtiles + waves_per_block - 1) / waves_per_block,
                                 32 * waves_per_block, 0, stream>>>(
            gacc, gcnt, Wo, bo, G, HID, NC, out);
    }
}